// CDFGPooling_2800318677628
// MI455X (gfx1250) — compile-verified
//
#include <hip/hip_runtime.h>
#include <hip/hip_bf16.h>
#include <math.h>

typedef __attribute__((ext_vector_type(2))) float v2f;
typedef __attribute__((ext_vector_type(8))) float v8f;

#define C_DIM 256
#define NPG   4096
#define BATCH 16
#define TOPK  1024

// ---------------------------------------------------------------------------
// K1: kqv = x @ w_kqv + b  via V_WMMA_F32_16X16X4_F32 (full fp32 precision).
// One wave handles a 16-node tile; 8 waves / block -> 128 nodes per block.
// B matrix = w_kqv (C x 3) zero-padded to (C x 16), staged in LDS as K-pairs:
//   wpad2[(K/2)*16 + n] = { B[K][n], B[K+1][n] }   (one ds_load_b64 / iter)
//
// VGPR layouts (CDNA5 ISA 7.12.2, 32-bit):
//   A 16x4 : lane L, vgpr v -> M = L%16, K = v + 2*(L/16)
//   B 4x16 : lane L, vgpr v -> N = L%16, K = v + 2*(L/16)
//   D 16x16: lane L, vgpr j -> N = L%16, M = j + 8*(L/16)
// ---------------------------------------------------------------------------
__global__ __launch_bounds__(256) void kqv_wmma_kernel(
    const float* __restrict__ x,      // Nn x 256
    const float* __restrict__ w,      // 256 x 3 (row-major)
    const float* __restrict__ b,      // 3
    float* __restrict__ k_out,
    float* __restrict__ q_out,
    float* __restrict__ v_out)
{
    __shared__ v2f wpad2[(C_DIM / 2) * 16];   // 16 KB
    const int tid = threadIdx.x;

    // Stage 1: zero the padded weight tile (uniform, no divergence).
    for (int i = tid; i < (C_DIM / 2) * 16; i += 256) {
        v2f z; z.x = 0.0f; z.y = 0.0f;
        wpad2[i] = z;
    }
    __syncthreads();
    // Stage 2: all 256 threads copy the 256x3 weights, 3 each (contiguous global reads).
    // element i -> (c = i/3, n = i%3); pair p = c/2, slot = c&1.
    for (int i = tid; i < C_DIM * 3; i += 256) {
        int c = i / 3, n = i - 3 * c;
        float wv = w[i];
        ((float*)&wpad2[(c >> 1) * 16 + n])[c & 1] = wv;
    }
    __syncthreads();

    const int wave = tid >> 5;
    const int lane = tid & 31;
    const int half = lane >> 4;   // 0 or 1
    const int ln   = lane & 15;
    const long tileBase = ((long)blockIdx.x * 8 + wave) * 16;
    const float* xrow = x + (tileBase + ln) * C_DIM;

    // Warm the row into cache (emits global_prefetch_b8).
    __builtin_prefetch(xrow + half * 128, 0, 3);

    // Two independent accumulator chains (even / odd K-steps).
    v8f acc0 = {};
    v8f acc1 = {};
    #pragma unroll 16
    for (int kk = 0; kk < C_DIM; kk += 8) {
        const int kA0 = kk + 2 * half;          // even step
        const int kA1 = kk + 4 + 2 * half;      // odd step
        v2f a0 = *(const v2f*)(xrow + kA0);
        v2f a1 = *(const v2f*)(xrow + kA1);
        v2f b0 = wpad2[(kA0 >> 1) * 16 + ln];
        v2f b1 = wpad2[(kA1 >> 1) * 16 + ln];
        acc0 = __builtin_amdgcn_wmma_f32_16x16x4_f32(
            false, a0, false, b0, (short)0, acc0, false, false);
        acc1 = __builtin_amdgcn_wmma_f32_16x16x4_f32(
            false, a1, false, b1, (short)0, acc1, false, false);
    }

    if (ln < 3) {
        const float bias = b[ln];
        float* dst = (ln == 0) ? k_out : (ln == 1) ? q_out : v_out;
        #pragma unroll
        for (int j = 0; j < 8; ++j) {
            const long row = tileBase + j + 8 * half;
            dst[row] = (acc0[j] + acc1[j]) + bias;
        }
    }
}

// ---------------------------------------------------------------------------
// Monotonic float <-> uint encoding for atomic segment-max.
// ---------------------------------------------------------------------------
__device__ __forceinline__ unsigned enc_f32(float f) {
    unsigned u = __float_as_uint(f);
    return (u & 0x80000000u) ? ~u : (u | 0x80000000u);
}
__device__ __forceinline__ float dec_f32(unsigned e) {
    unsigned b = (e & 0x80000000u) ? (e & 0x7FFFFFFFu) : ~e;
    return __uint_as_float(b);
}

__global__ void init_kernel(unsigned* __restrict__ m_u,
                            float* __restrict__ den,
                            float* __restrict__ num, int n2)
{
    int i = blockIdx.x * blockDim.x + threadIdx.x;
    if (i >= n2) return;
    m_u[i] = 0u;            // encodes "below -inf" (no edge seen)
    den[i] = 0.0f;
    num[i] = 0.0f;
}

// K2: per-edge segment max of logits (pass 1)
__global__ void edge_max_kernel(
    const int* __restrict__ ei, int E,
    const float* __restrict__ ksrc, const float* __restrict__ qdst,
    const float* __restrict__ krel_w, const float* __restrict__ krel_b,
    const float* __restrict__ p_rel, int t, int dstOff,
    unsigned* __restrict__ m_u)
{
    int e = blockIdx.x * blockDim.x + threadIdx.x;
    if (e >= E) return;
    int s = ei[e];
    int d = ei[E + e];
    float ke    = ksrc[s] * krel_w[t] + krel_b[t];
    float logit = qdst[d] * ke * p_rel[t];
    atomicMax(&m_u[d + dstOff], enc_f32(logit));
}

// K3: per-edge exp/sum (pass 2): den += ex, num += ex * ve
__global__ void edge_sum_kernel(
    const int* __restrict__ ei, int E,
    const float* __restrict__ ksrc, const float* __restrict__ vsrc,
    const float* __restrict__ qdst,
    const float* __restrict__ krel_w, const float* __restrict__ krel_b,
    const float* __restrict__ vrel_w, const float* __restrict__ vrel_b,
    const float* __restrict__ p_rel, int t, int dstOff,
    const unsigned* __restrict__ m_u,
    float* __restrict__ den, float* __restrict__ num)
{
    int e = blockIdx.x * blockDim.x + threadIdx.x;
    if (e >= E) return;
    int s = ei[e];
    int d = ei[E + e];
    float ke    = ksrc[s] * krel_w[t] + krel_b[t];
    float ve    = vsrc[s] * vrel_w[t] + vrel_b[t];
    float logit = qdst[d] * ke * p_rel[t];
    int dg = d + dstOff;
    unsigned u = m_u[dg];
    float m = 0.0f;
    if (u != 0u) {
        m = dec_f32(u);
        if (!isfinite(m)) m = 0.0f;   // matches jnp.where(isfinite(m), m, 0)
    }
    float ex = expf(logit - m);
    atomicAdd(&den[dg], ex);
    atomicAdd(&num[dg], ex * ve);
}

// K4: a = num/den ; s = gelu(a*w_out + b_out)  (exact erf gelu)
__global__ void score_kernel(
    const float* __restrict__ den, const float* __restrict__ num,
    const float* __restrict__ w_out_i, const float* __restrict__ b_out_i,
    const float* __restrict__ w_out_d, const float* __restrict__ b_out_d,
    float* __restrict__ s, int Nn)
{
    int i = blockIdx.x * blockDim.x + threadIdx.x;
    if (i >= 2 * Nn) return;
    float dd = den[i];
    float a  = (dd != 0.0f) ? (num[i] / dd) : 0.0f;
    float w, bb;
    if (i < Nn) { w = w_out_i[0]; bb = b_out_i[0]; }
    else        { w = w_out_d[0]; bb = b_out_d[0]; }
    float z = a * w + bb;
    s[i] = 0.5f * z * (1.0f + erff(z * 0.70710678118654752440f));
}

// ---------------------------------------------------------------------------
// K5: per (batch, matrix): bitonic top-K (K=1024 of 4096) in LDS, then
// tanh-weighted mean of the selected x rows -> out[b, mat*256 + c].
// ---------------------------------------------------------------------------
__global__ __launch_bounds__(256) void pool_kernel(
    const float* __restrict__ x_inst, const float* __restrict__ x_data,
    const float* __restrict__ s, float* __restrict__ out, int Nn)
{
    __shared__ float vals[NPG];
    __shared__ int   idxs[NPG];
    __shared__ float wk[TOPK];

    const int b   = blockIdx.x;
    const int mat = blockIdx.y;
    const int tid = threadIdx.x;
    const float* x  = mat ? x_data : x_inst;
    const float* sv = s + (size_t)mat * Nn + (size_t)b * NPG;

    for (int i = tid; i < NPG; i += 256) { vals[i] = sv[i]; idxs[i] = i; }
    __syncthreads();

    // Bitonic sort, descending by value, carrying indices.
    for (int size = 2; size <= NPG; size <<= 1) {
        for (int stride = size >> 1; stride > 0; stride >>= 1) {
            for (int i = tid; i < NPG; i += 256) {
                int j = i ^ stride;
                if (j > i) {
                    bool desc = ((i & size) == 0);
                    float vi = vals[i], vj = vals[j];
                    bool doswap = desc ? (vi < vj) : (vi > vj);
                    if (doswap) {
                        vals[i] = vj; vals[j] = vi;
                        int t = idxs[i]; idxs[i] = idxs[j]; idxs[j] = t;
                    }
                }
            }
            __syncthreads();
        }
    }

    for (int k = tid; k < TOPK; k += 256) wk[k] = tanhf(vals[k]);
    __syncthreads();

    const int c = tid;   // 0..255
    float acc = 0.0f;
    for (int k = 0; k < TOPK; ++k) {
        int node = b * NPG + idxs[k];
        acc += x[(size_t)node * C_DIM + c] * wk[k];
    }
    out[b * (2 * C_DIM) + mat * C_DIM + c] = acc * (1.0f / (float)TOPK);
}

// ---------------------------------------------------------------------------
extern "C" void kernel_launch(void* const* d_in, const int* in_sizes, int n_in,
                              void* d_out, int out_size, void* d_ws, size_t ws_size,
                              hipStream_t stream)
{
    const float* x_inst     = (const float*)d_in[0];
    const float* x_data     = (const float*)d_in[1];
    const float* w_kqv_inst = (const float*)d_in[2];
    const float* b_kqv_inst = (const float*)d_in[3];
    const float* w_kqv_data = (const float*)d_in[4];
    const float* b_kqv_data = (const float*)d_in[5];
    const float* w_out_inst = (const float*)d_in[6];
    const float* b_out_inst = (const float*)d_in[7];
    const float* w_out_data = (const float*)d_in[8];
    const float* b_out_data = (const float*)d_in[9];
    const float* krel_w     = (const float*)d_in[10];
    const float* krel_b     = (const float*)d_in[11];
    const float* vrel_w     = (const float*)d_in[12];
    const float* vrel_b     = (const float*)d_in[13];
    const float* p_rel      = (const float*)d_in[14];
    const int*   ei_i2d     = (const int*)d_in[15];
    const int*   ei_d2i     = (const int*)d_in[16];
    const int*   ei_i2i     = (const int*)d_in[17];

    const int Nn = in_sizes[0] / C_DIM;      // 65536
    const int E  = in_sizes[15] / 2;         // 1048576

    // Workspace layout (floats): k_i,q_i,v_i,k_d,q_d,v_d [6N], m_u [2N],
    // den [2N], num [2N], s [2N]  => 14N floats ~ 3.5 MB
    float* ws  = (float*)d_ws;
    float* k_i = ws;
    float* q_i = k_i + Nn;
    float* v_i = q_i + Nn;
    float* k_d = v_i + Nn;
    float* q_d = k_d + Nn;
    float* v_d = q_d + Nn;
    unsigned* m_u = (unsigned*)(v_d + Nn);        // 2N
    float* den = (float*)(m_u + 2 * Nn);          // 2N
    float* num = den + 2 * Nn;                    // 2N
    float* sArr = num + 2 * Nn;                   // 2N

    // K1: kqv projections via fp32 WMMA (128 nodes per block)
    const int kqvBlocks = Nn / 128;
    kqv_wmma_kernel<<<kqvBlocks, 256, 0, stream>>>(x_inst, w_kqv_inst, b_kqv_inst, k_i, q_i, v_i);
    kqv_wmma_kernel<<<kqvBlocks, 256, 0, stream>>>(x_data, w_kqv_data, b_kqv_data, k_d, q_d, v_d);

    // init segment buffers
    init_kernel<<<(2 * Nn + 255) / 256, 256, 0, stream>>>(m_u, den, num, 2 * Nn);

    const int eb = (E + 255) / 256;
    // K2: segment max   (t0: i2d src=inst, q=q_d, off=N ; t1: d2i ; t2: i2i)
    edge_max_kernel<<<eb, 256, 0, stream>>>(ei_i2d, E, k_i, q_d, krel_w, krel_b, p_rel, 0, Nn, m_u);
    edge_max_kernel<<<eb, 256, 0, stream>>>(ei_d2i, E, k_d, q_i, krel_w, krel_b, p_rel, 1, 0,  m_u);
    edge_max_kernel<<<eb, 256, 0, stream>>>(ei_i2i, E, k_i, q_i, krel_w, krel_b, p_rel, 2, 0,  m_u);

    // K3: exp / segment sums
    edge_sum_kernel<<<eb, 256, 0, stream>>>(ei_i2d, E, k_i, v_i, q_d, krel_w, krel_b, vrel_w, vrel_b, p_rel, 0, Nn, m_u, den, num);
    edge_sum_kernel<<<eb, 256, 0, stream>>>(ei_d2i, E, k_d, v_d, q_i, krel_w, krel_b, vrel_w, vrel_b, p_rel, 1, 0,  m_u, den, num);
    edge_sum_kernel<<<eb, 256, 0, stream>>>(ei_i2i, E, k_i, v_i, q_i, krel_w, krel_b, vrel_w, vrel_b, p_rel, 2, 0,  m_u, den, num);

    // K4: scores
    score_kernel<<<(2 * Nn + 255) / 256, 256, 0, stream>>>(den, num, w_out_inst, b_out_inst,
                                                           w_out_data, b_out_data, sArr, Nn);

    // K5: top-K + tanh-weighted pooling -> (16, 512)
    dim3 pg(BATCH, 2);
    pool_kernel<<<pg, 256, 0, stream>>>(x_inst, x_data, sArr, (float*)d_out, Nn);
}